// AttrAttentionLayer_52587579572318
// MI455X (gfx1250) — compile-verified
//
#include <hip/hip_runtime.h>

// GAT-style attention layer, f32 end-to-end, WMMA f32 16x16x4 for both GEMMs.
// Memory-bound on adj + node_type (2 x 268 MB reads ~ 23 us @ 23.3 TB/s).
// B operands for WMMA are kept in a K-paired layout so each lane's (K,K+1)
// fragment is one 8-byte load straight into an even-aligned VGPR pair.

#define LRELU_ALPHA 0.2f
#define NEG_INF_F  -9000000000000000.0f

constexpr int B    = 8;
constexpr int N    = 2048;
constexpr int FIN  = 256;
constexpr int FOUT = 64;
constexpr int SPLIT  = 8;   // i-split for column stats
constexpr int SPLITJ = 2;   // j-split for h' accumulation

// Paired-W LDS row stride (dwords) for k2-row k/2: 160 = 128 data + 32 pad.
// 160 % 64 == 32 -> the lanes>=16 K-group (k2+1 row) lands on the opposite
// 32-bank half from lanes<16 -> conflict-free ds_load_b64.
constexpr int PSTR = 160;

typedef __attribute__((ext_vector_type(2))) float v2f;
typedef __attribute__((ext_vector_type(8))) float v8f;

// ---------------------------------------------------------------------------
// k1: Wh = h @ W  (WMMA f32 16x16x4), fused f1/f2 and Whl = Wh*level epilogue.
// Grid: (N/(16*8), 1, B), block 256 (8 waves, one 16-row tile each).
// ---------------------------------------------------------------------------
__global__ __launch_bounds__(256)
void k1_wh(const float* __restrict__ h, const float* __restrict__ W,
           const float* __restrict__ avec, const float* __restrict__ level,
           float* __restrict__ whlp, float* __restrict__ f1, float* __restrict__ f2)
{
    __shared__ float sWp[(FIN / 2) * PSTR];     // 80 KB, K-paired W
    const int tid = threadIdx.x;
    for (int t = tid; t < FIN * FOUT; t += 256) {
        const int k = t / FOUT, col = t % FOUT;
        sWp[(k >> 1) * PSTR + col * 2 + (k & 1)] = W[t];
    }
    __syncthreads();

    const int wave = tid >> 5, lane = tid & 31;
    const int lm = lane & 15, lh = lane >> 4;
    const int b  = blockIdx.z;
    const int i0 = (blockIdx.x * 8 + wave) * 16;

    v8f c[4] = {};
    // A tile (16x4 f32): lanes 0-15 hold M=lane, K={k0,k0+1}; lanes 16-31 K={k0+2,k0+3}
    const float* hrow = h + ((long)(b * N + i0 + lm)) * FIN + lh * 2;
    for (int k0 = 0; k0 < FIN; k0 += 4) {
        const v2f av = *reinterpret_cast<const v2f*>(hrow + k0);
        const float* wrow = sWp + ((k0 >> 1) + lh) * PSTR;   // this lane's k2-row
#pragma unroll
        for (int t = 0; t < 4; ++t) {
            const int col = t * 16 + lm;
            const v2f bv = *reinterpret_cast<const v2f*>(wrow + col * 2);
            c[t] = __builtin_amdgcn_wmma_f32_16x16x4_f32(
                false, av, false, bv, (short)0, c[t], false, false);
        }
    }

    // f1 = Wh@a1, f2 = Wh@a2 : per-lane partials then 16-lane xor reduction.
    float a1c[4], a2c[4];
#pragma unroll
    for (int t = 0; t < 4; ++t) {
        a1c[t] = avec[t * 16 + lm];
        a2c[t] = avec[FOUT + t * 16 + lm];
    }
#pragma unroll
    for (int v = 0; v < 8; ++v) {
        float s1 = 0.f, s2 = 0.f;
#pragma unroll
        for (int t = 0; t < 4; ++t) { s1 += c[t][v] * a1c[t]; s2 += c[t][v] * a2c[t]; }
#pragma unroll
        for (int m = 1; m < 16; m <<= 1) {
            s1 += __shfl_xor(s1, m, 32);
            s2 += __shfl_xor(s2, m, 32);
        }
        if (lm == 0) {                            // lanes 0 and 16
            const int row = i0 + v + lh * 8;      // C layout: VGPRv -> M=v / M=v+8
            f1[b * N + row] = s1;
            f2[b * N + row] = s2;
        }
    }

    // Whl = Wh * level, stored K-paired: whlp[(b*N+row)/2][col*2 + row%2]
#pragma unroll
    for (int v = 0; v < 8; ++v) {
        const int row = i0 + v + lh * 8;
        const float lv = level[b * N + row];
        const long jp = (long)(b * N + row) >> 1;
        const int  par = row & 1;
#pragma unroll
        for (int t = 0; t < 4; ++t) {
            const int col = t * 16 + lm;
            whlp[jp * (2 * FOUT) + col * 2 + par] = c[t][v] * lv;
        }
    }
}

// ---------------------------------------------------------------------------
// k2: per-column (axis=1 softmax) online max / sum-exp, i-split partials.
// Grid: (N/256, SPLIT, B), block 256. Coalesced in j.
// ---------------------------------------------------------------------------
__global__ __launch_bounds__(256)
void k2_stats(const int* __restrict__ adj, const float* __restrict__ nt,
              const float* __restrict__ f1, const float* __restrict__ f2,
              float* __restrict__ partM, float* __restrict__ partS)
{
    const int b = blockIdx.z;
    const int j = blockIdx.x * 256 + threadIdx.x;
    const int iBeg = blockIdx.y * (N / SPLIT);
    const float f2j = f2[b * N + j];
    const long base = ((long)b * N) * N + j;

    float m = -__builtin_inff(), s = 0.f;
#pragma unroll 4
    for (int i = iBeg; i < iBeg + N / SPLIT; ++i) {
        const float f1i = f1[b * N + i];          // uniform -> scalar path
        float e = f1i + f2j;
        e = e > 0.f ? e : LRELU_ALPHA * e;
        const long o = base + (long)i * N;
        const float x = (adj[o] > 0) ? e * nt[o] : NEG_INF_F;
        const float mn = fmaxf(m, x);
        s = s * __expf(m - mn) + __expf(x - mn);
        m = mn;
    }
    const long po = ((long)b * SPLIT + blockIdx.y) * N + j;
    partM[po] = m;
    partS[po] = s;
}

__global__ __launch_bounds__(256)
void k2b_combine(const float* __restrict__ partM, const float* __restrict__ partS,
                 float* __restrict__ colM, float* __restrict__ colS)
{
    const int idx = blockIdx.x * 256 + threadIdx.x;   // b*N + j
    const int b = idx / N, j = idx % N;
    float m = -__builtin_inff();
#pragma unroll
    for (int k = 0; k < SPLIT; ++k)
        m = fmaxf(m, partM[((long)b * SPLIT + k) * N + j]);
    float s = 0.f;
#pragma unroll
    for (int k = 0; k < SPLIT; ++k)
        s += partS[((long)b * SPLIT + k) * N + j] *
             __expf(partM[((long)b * SPLIT + k) * N + j] - m);
    colM[idx] = m;
    colS[idx] = s;
}

// ---------------------------------------------------------------------------
// k3: h'_part = P @ Whl, P computed on the fly, WMMA f32 16x16x4.
// Grid: (N/(16*4), SPLITJ, B), block 128 (4 waves, 16 rows each).
// Each wave: loop 64-wide j panels; stage P(16x64) in LDS (stride 68 -> no
// bank conflicts on A-tile reads), then 16 WMMA K-steps x 4 N-tiles with
// single-b64 B-fragment loads from the K-paired Whl.
// ---------------------------------------------------------------------------
#define WAVES3 4
__global__ __launch_bounds__(WAVES3 * 32)
void k3_hprime(const int* __restrict__ adj, const float* __restrict__ nt,
               const float* __restrict__ f1, const float* __restrict__ f2,
               const float* __restrict__ colM, const float* __restrict__ colS,
               const float* __restrict__ whlp, float* __restrict__ hpPart)
{
    __shared__ float sP[WAVES3 * 16 * 68];       // ~17.4 KB
    const int tid = threadIdx.x, wave = tid >> 5, lane = tid & 31;
    const int lm = lane & 15, lh = lane >> 4;
    const int b  = blockIdx.z;
    const int i0 = (blockIdx.x * WAVES3 + wave) * 16;
    const int jBeg = blockIdx.y * (N / SPLITJ);
    float* P = sP + wave * 16 * 68;

    v8f c[4] = {};
    const int jA = 2 * lane;                     // lane's j pair within panel
    for (int j0 = jBeg; j0 < jBeg + N / SPLITJ; j0 += 64) {
        const int ja = j0 + jA;
        const float f2a = f2[b * N + ja],     f2b = f2[b * N + ja + 1];
        const float Ma  = colM[b * N + ja],   Mb  = colM[b * N + ja + 1];
        const float rSa = 1.f / colS[b * N + ja];
        const float rSb = 1.f / colS[b * N + ja + 1];

        for (int r = 0; r < 16; ++r) {
            const int row = i0 + r;
            const float f1r = f1[b * N + row];
            const long base = ((long)(b * N + row)) * N + ja;
            const int2 ad = *reinterpret_cast<const int2*>(adj + base);
            const v2f ntv = *reinterpret_cast<const v2f*>(nt + base);
            if (j0 + 64 < jBeg + N / SPLITJ) {   // prefetch next panel's row
                __builtin_prefetch(adj + base + 64, 0, 0);
                __builtin_prefetch(nt + base + 64, 0, 0);
            }
            float ea = f1r + f2a; ea = ea > 0.f ? ea : LRELU_ALPHA * ea;
            float eb = f1r + f2b; eb = eb > 0.f ? eb : LRELU_ALPHA * eb;
            const float xa = (ad.x > 0) ? ea * ntv.x : NEG_INF_F;
            const float xb = (ad.y > 0) ? eb * ntv.y : NEG_INF_F;
            v2f pv;
            pv.x = __expf(xa - Ma) * rSa;
            pv.y = __expf(xb - Mb) * rSb;
            *reinterpret_cast<v2f*>(P + r * 68 + jA) = pv;
        }
        // (per-wave LDS region; same-wave DS ops are in-order -> no barrier)
#pragma unroll 4
        for (int ks = 0; ks < 16; ++ks) {
            const int k0 = ks * 4;
            const v2f av = *reinterpret_cast<const v2f*>(P + lm * 68 + k0 + lh * 2);
            // K-pair base for this lane's K-group: (b*N + j0 + k0)/2 + lh
            const float* wb = whlp +
                (((long)(b * N + j0 + k0) >> 1) + lh) * (2 * FOUT);
#pragma unroll
            for (int t = 0; t < 4; ++t) {
                const int col = t * 16 + lm;
                const v2f bv = *reinterpret_cast<const v2f*>(wb + col * 2);
                c[t] = __builtin_amdgcn_wmma_f32_16x16x4_f32(
                    false, av, false, bv, (short)0, c[t], false, false);
            }
        }
    }

    float* part = hpPart + (long)blockIdx.y * B * N * FOUT;
#pragma unroll
    for (int v = 0; v < 8; ++v) {
        const int row = i0 + v + lh * 8;
#pragma unroll
        for (int t = 0; t < 4; ++t)
            part[((long)(b * N + row)) * FOUT + t * 16 + lm] = c[t][v];
    }
}

// k4: out = relu(part0 + part1)
__global__ __launch_bounds__(256)
void k4_combine(const float* __restrict__ hpPart, float* __restrict__ out)
{
    const long idx = (long)blockIdx.x * 256 + threadIdx.x;
    out[idx] = fmaxf(hpPart[idx] + hpPart[(long)B * N * FOUT + idx], 0.f);
}

// ---------------------------------------------------------------------------
extern "C" void kernel_launch(void* const* d_in, const int* in_sizes, int n_in,
                              void* d_out, int out_size, void* d_ws, size_t ws_size,
                              hipStream_t stream)
{
    const float* h         = (const float*)d_in[0];
    const int*   adj       = (const int*)d_in[1];
    const float* level     = (const float*)d_in[2];
    const float* node_type = (const float*)d_in[3];
    const float* W         = (const float*)d_in[4];
    const float* avec      = (const float*)d_in[5];
    float* out = (float*)d_out;

    // workspace layout (floats); ~14 MB total
    float* w      = (float*)d_ws;
    float* whlp   = w;                       // B*N*FOUT   = 1048576 (K-paired)
    float* f1     = w + 1048576;             // B*N        = 16384
    float* f2     = f1 + 16384;
    float* colM   = f2 + 16384;
    float* colS   = colM + 16384;
    float* partM  = colS + 16384;            // B*SPLIT*N  = 131072
    float* partS  = partM + 131072;
    float* hpPart = partS + 131072;          // SPLITJ*B*N*FOUT = 2097152

    k1_wh<<<dim3(N / (16 * 8), 1, B), 256, 0, stream>>>(h, W, avec, level, whlp, f1, f2);
    k2_stats<<<dim3(N / 256, SPLIT, B), 256, 0, stream>>>(adj, node_type, f1, f2, partM, partS);
    k2b_combine<<<dim3(B * N / 256), 256, 0, stream>>>(partM, partS, colM, colS);
    k3_hprime<<<dim3(N / (16 * WAVES3), SPLITJ, B), WAVES3 * 32, 0, stream>>>(
        adj, node_type, f1, f2, colM, colS, whlp, hpPart);
    k4_combine<<<dim3(B * N * FOUT / 256), 256, 0, stream>>>(hpPart, out);
}